// BetterModel_22557168239477
// MI455X (gfx1250) — compile-verified
//
#include <hip/hip_runtime.h>
#include <math.h>

typedef __attribute__((ext_vector_type(2))) float v2f;
typedef __attribute__((ext_vector_type(8))) float v8f;

#define N_NODES  50000
#define N_EDGES  1600000
#define IN_DIM   128
#define HIDDEN   256
#define N_GRAPHS 512
#define BN_EPS   1e-5f

// ---------------------------------------------------------------- utilities
__global__ void zero_kernel(float* p, int n) {
    int i = blockIdx.x * blockDim.x + threadIdx.x;
    if (i < n) p[i] = 0.f;
}

// Hin = x  (float4 copy; 50000*128 = 6.4M floats = 1.6M float4)
__global__ void copy_x_kernel(const float* __restrict__ x, float* __restrict__ Hin) {
    int i = blockIdx.x * blockDim.x + threadIdx.x;   // 1.6M float4
    ((float4*)Hin)[i] = ((const float4*)x)[i];
}

// Hin[dst] += x[src] : one wave per edge, lane l handles channels 4l..4l+3
__global__ void scatter_add_kernel(const float* __restrict__ x,
                                   const int* __restrict__ ei,
                                   float* __restrict__ Hin) {
    int gid = blockIdx.x * blockDim.x + threadIdx.x;     // < 51.2M
    int e = gid >> 5;
    int c = (gid & 31) * 4;
    int src = ei[e];
    int dst = ei[N_EDGES + e];
    float4 v = *(const float4*)(x + (size_t)src * IN_DIM + c);
    float* p = Hin + (size_t)dst * IN_DIM + c;
    atomicAdd(p + 0, v.x);
    atomicAdd(p + 1, v.y);
    atomicAdd(p + 2, v.z);
    atomicAdd(p + 3, v.w);
}

// ------------------------------------------------- GEMM1: H1 = Hin@W1 + b1
// block = 128 threads (4 waves); block owns 16 rows x 256 cols.
// wave w owns cols [64w, 64w+64). f32 WMMA 16x16x4, K = 128.
// Epilogue also accumulates per-column sum / sumsq for BN1.
__global__ void gemm1_wmma_kernel(const float* __restrict__ Hin,
                                  const float* __restrict__ W1,
                                  const float* __restrict__ b1,
                                  float* __restrict__ H1,
                                  float* __restrict__ colsum,
                                  float* __restrict__ colsumsq) {
    const int lane  = threadIdx.x & 31;
    const int wave  = threadIdx.x >> 5;
    const int half  = lane >> 4;          // 0: lanes 0-15, 1: lanes 16-31
    const int laneN = lane & 15;
    const int mBase = blockIdx.x * 16;
    const int nBase = wave * 64;
    const int m     = mBase + laneN;      // A row held by this lane

    v8f acc[4] = {};
    const float* aRow = Hin + (size_t)m * IN_DIM + 2 * half;

    for (int k = 0; k < IN_DIM; k += 4) {
        v2f a = *(const v2f*)(aRow + k);  // A[m][k+2h], A[m][k+2h+1]
        #pragma unroll
        for (int j = 0; j < 4; ++j) {
            int col = nBase + j * 16 + laneN;
            v2f b;
            b.x = W1[(size_t)(k + 2 * half + 0) * HIDDEN + col];
            b.y = W1[(size_t)(k + 2 * half + 1) * HIDDEN + col];
            acc[j] = __builtin_amdgcn_wmma_f32_16x16x4_f32(
                false, a, false, b, (short)0, acc[j], false, false);
        }
    }

    const int row0 = mBase + 8 * half;    // C layout: lane-half => rows +8
    #pragma unroll
    for (int j = 0; j < 4; ++j) {
        int col = nBase + j * 16 + laneN;
        float bias = b1[col];
        float s = 0.f, sq = 0.f;
        #pragma unroll
        for (int r = 0; r < 8; ++r) {
            float v = acc[j][r] + bias;
            H1[(size_t)(row0 + r) * HIDDEN + col] = v;
            s += v;
            sq += v * v;
        }
        s  += __shfl_xor(s, 16, 32);      // combine the two row halves (same col)
        sq += __shfl_xor(sq, 16, 32);
        if (half == 0) {
            atomicAdd(&colsum[col], s);
            atomicAdd(&colsumsq[col], sq);
        }
    }
}

// BN1 -> affine form:  relu_in = H1*scale1 + shift1
__global__ void bn1_finalize_kernel(const float* __restrict__ colsum,
                                    const float* __restrict__ colsumsq,
                                    const float* __restrict__ g1,
                                    const float* __restrict__ be1,
                                    float* __restrict__ scale1,
                                    float* __restrict__ shift1) {
    int c = threadIdx.x;
    float mu  = colsum[c] * (1.f / N_NODES);
    float var = colsumsq[c] * (1.f / N_NODES) - mu * mu;
    float sc  = g1[c] * rsqrtf(var + BN_EPS);
    scale1[c] = sc;
    shift1[c] = be1[c] - mu * sc;
}

// ---------------------- GEMM2: H2 = relu( relu(BN(H1)) @ W2 + b2 ), K = 256
__global__ void gemm2_wmma_kernel(const float* __restrict__ H1,
                                  const float* __restrict__ scale1,
                                  const float* __restrict__ shift1,
                                  const float* __restrict__ W2,
                                  const float* __restrict__ b2,
                                  float* __restrict__ H2) {
    const int lane  = threadIdx.x & 31;
    const int wave  = threadIdx.x >> 5;
    const int half  = lane >> 4;
    const int laneN = lane & 15;
    const int mBase = blockIdx.x * 16;
    const int nBase = wave * 64;
    const int m     = mBase + laneN;

    v8f acc[4] = {};
    const float* aRow = H1 + (size_t)m * HIDDEN + 2 * half;

    for (int k = 0; k < HIDDEN; k += 4) {
        int ka = k + 2 * half;
        v2f h = *(const v2f*)(aRow + k);
        v2f a;
        a.x = fmaxf(h.x * scale1[ka] + shift1[ka], 0.f);          // BN + ReLU
        a.y = fmaxf(h.y * scale1[ka + 1] + shift1[ka + 1], 0.f);
        #pragma unroll
        for (int j = 0; j < 4; ++j) {
            int col = nBase + j * 16 + laneN;
            v2f b;
            b.x = W2[(size_t)(ka + 0) * HIDDEN + col];
            b.y = W2[(size_t)(ka + 1) * HIDDEN + col];
            acc[j] = __builtin_amdgcn_wmma_f32_16x16x4_f32(
                false, a, false, b, (short)0, acc[j], false, false);
        }
    }

    const int row0 = mBase + 8 * half;
    #pragma unroll
    for (int j = 0; j < 4; ++j) {
        int col = nBase + j * 16 + laneN;
        float bias = b2[col];
        #pragma unroll
        for (int r = 0; r < 8; ++r) {
            float v = fmaxf(acc[j][r] + bias, 0.f);               // outer ReLU
            H2[(size_t)(row0 + r) * HIDDEN + col] = v;
        }
    }
}

// gate[i] = H2[i,:] . Wg + bg   (1 wave per node)
__global__ void gate_kernel(const float* __restrict__ H2,
                            const float* __restrict__ Wg,
                            const float* __restrict__ bg,
                            float* __restrict__ gate) {
    int lane = threadIdx.x & 31;
    int wave = threadIdx.x >> 5;
    int node = blockIdx.x * 8 + wave;
    if (node >= N_NODES) return;
    const float* row = H2 + (size_t)node * HIDDEN;
    float acc = 0.f;
    #pragma unroll
    for (int j = 0; j < 8; ++j) {
        int c = lane + 32 * j;
        acc += row[c] * Wg[c];
    }
    #pragma unroll
    for (int o = 16; o > 0; o >>= 1) acc += __shfl_xor(acc, o, 32);
    if (lane == 0) gate[node] = acc + bg[0];
}

// segment softmax + weighted pooling: 1 block per graph (batch is sorted)
__global__ void pool_kernel(const float* __restrict__ H2,
                            const float* __restrict__ gate,
                            const int* __restrict__ batch,
                            float* __restrict__ pooled) {
    __shared__ float red[256];
    __shared__ int   sRange[2];
    const int g = blockIdx.x, tid = threadIdx.x;

    if (tid == 0) {
        int lo = 0, hi = N_NODES;                     // lower_bound(batch, g)
        while (lo < hi) { int mid = (lo + hi) >> 1; if (batch[mid] < g) lo = mid + 1; else hi = mid; }
        sRange[0] = lo;
        lo = 0; hi = N_NODES;                         // lower_bound(batch, g+1)
        while (lo < hi) { int mid = (lo + hi) >> 1; if (batch[mid] < g + 1) lo = mid + 1; else hi = mid; }
        sRange[1] = lo;
    }
    __syncthreads();
    const int start = sRange[0], end = sRange[1];

    // segment max
    float lmax = -3.402823466e38f;
    for (int i = start + tid; i < end; i += 256) lmax = fmaxf(lmax, gate[i]);
    red[tid] = lmax; __syncthreads();
    for (int s = 128; s > 0; s >>= 1) {
        if (tid < s) red[tid] = fmaxf(red[tid], red[tid + s]);
        __syncthreads();
    }
    const float m = red[0]; __syncthreads();

    // segment sum of exp
    float lsum = 0.f;
    for (int i = start + tid; i < end; i += 256) lsum += expf(gate[i] - m);
    red[tid] = lsum; __syncthreads();
    for (int s = 128; s > 0; s >>= 1) {
        if (tid < s) red[tid] += red[tid + s];
        __syncthreads();
    }
    const float inv = (end > start) ? 1.f / red[0] : 0.f;

    // weighted sum: thread tid owns channel tid
    float acc = 0.f;
    for (int i = start; i < end; ++i) {
        float w = expf(gate[i] - m);
        acc += w * H2[(size_t)i * HIDDEN + tid];
    }
    pooled[(size_t)g * HIDDEN + tid] = acc * inv;
}

// BN2 over the 512 pooled rows -> affine scale2/shift2
__global__ void bn2_stats_kernel(const float* __restrict__ pooled,
                                 const float* __restrict__ g2,
                                 const float* __restrict__ be2,
                                 float* __restrict__ scale2,
                                 float* __restrict__ shift2) {
    int c = threadIdx.x;
    float s = 0.f, sq = 0.f;
    for (int g = 0; g < N_GRAPHS; ++g) {
        float v = pooled[(size_t)g * HIDDEN + c];
        s += v; sq += v * v;
    }
    float mu  = s * (1.f / N_GRAPHS);
    float var = sq * (1.f / N_GRAPHS) - mu * mu;
    float sc  = g2[c] * rsqrtf(var + BN_EPS);
    scale2[c] = sc;
    shift2[c] = be2[c] - mu * sc;
}

// head: z = BN(pooled) @ Wh + bh ; out = log_softmax(z)   (1 block per graph)
__global__ void head_kernel(const float* __restrict__ pooled,
                            const float* __restrict__ scale2,
                            const float* __restrict__ shift2,
                            const float* __restrict__ Wh,
                            const float* __restrict__ bh,
                            float* __restrict__ out) {
    __shared__ float r0[256], r1[256];
    const int g = blockIdx.x, t = threadIdx.x;
    float zn = pooled[(size_t)g * HIDDEN + t] * scale2[t] + shift2[t];
    r0[t] = zn * Wh[t * 2 + 0];
    r1[t] = zn * Wh[t * 2 + 1];
    __syncthreads();
    for (int s = 128; s > 0; s >>= 1) {
        if (t < s) { r0[t] += r0[t + s]; r1[t] += r1[t + s]; }
        __syncthreads();
    }
    if (t == 0) {
        float z0 = r0[0] + bh[0];
        float z1 = r1[0] + bh[1];
        float mm = fmaxf(z0, z1);
        float lse = mm + logf(expf(z0 - mm) + expf(z1 - mm));
        out[g * 2 + 0] = z0 - lse;
        out[g * 2 + 1] = z1 - lse;
    }
}

// ---------------------------------------------------------------- launcher
extern "C" void kernel_launch(void* const* d_in, const int* in_sizes, int n_in,
                              void* d_out, int out_size, void* d_ws, size_t ws_size,
                              hipStream_t stream) {
    const float* x   = (const float*)d_in[0];
    const int*   ei  = (const int*)d_in[1];
    const int*   bat = (const int*)d_in[2];
    const float* W1  = (const float*)d_in[3];
    const float* b1  = (const float*)d_in[4];
    const float* g1  = (const float*)d_in[5];
    const float* be1 = (const float*)d_in[6];
    const float* W2  = (const float*)d_in[7];
    const float* b2  = (const float*)d_in[8];
    const float* Wg  = (const float*)d_in[9];
    const float* bg  = (const float*)d_in[10];
    const float* g2  = (const float*)d_in[11];
    const float* be2 = (const float*)d_in[12];
    const float* Wh  = (const float*)d_in[13];
    const float* bh  = (const float*)d_in[14];
    float* out = (float*)d_out;

    float* ws       = (float*)d_ws;
    float* Hin      = ws;                      // 50000*128
    float* H1       = Hin + (size_t)N_NODES * IN_DIM;   // 50000*256
    float* H2       = H1 + (size_t)N_NODES * HIDDEN;    // 50000*256
    float* gate     = H2 + (size_t)N_NODES * HIDDEN;    // 50000
    float* colsum   = gate + N_NODES;          // 256
    float* colsumsq = colsum + HIDDEN;         // 256
    float* scale1   = colsumsq + HIDDEN;       // 256
    float* shift1   = scale1 + HIDDEN;         // 256
    float* pooled   = shift1 + HIDDEN;         // 512*256
    float* scale2   = pooled + (size_t)N_GRAPHS * HIDDEN;
    float* shift2   = scale2 + HIDDEN;

    zero_kernel<<<2, 256, 0, stream>>>(colsum, 2 * HIDDEN);
    copy_x_kernel<<<(N_NODES * IN_DIM / 4) / 256, 256, 0, stream>>>(x, Hin);
    scatter_add_kernel<<<(N_EDGES * 32) / 256, 256, 0, stream>>>(x, ei, Hin);
    gemm1_wmma_kernel<<<N_NODES / 16, 128, 0, stream>>>(Hin, W1, b1, H1, colsum, colsumsq);
    bn1_finalize_kernel<<<1, 256, 0, stream>>>(colsum, colsumsq, g1, be1, scale1, shift1);
    gemm2_wmma_kernel<<<N_NODES / 16, 128, 0, stream>>>(H1, scale1, shift1, W2, b2, H2);
    gate_kernel<<<(N_NODES + 7) / 8, 256, 0, stream>>>(H2, Wg, bg, gate);
    pool_kernel<<<N_GRAPHS, 256, 0, stream>>>(H2, gate, bat, pooled);
    bn2_stats_kernel<<<1, 256, 0, stream>>>(pooled, g2, be2, scale2, shift2);
    head_kernel<<<N_GRAPHS, 256, 0, stream>>>(pooled, scale2, shift2, Wh, bh, out);
}